// Multi_View_Attention_69526930588481
// MI455X (gfx1250) — compile-verified
//
#include <hip/hip_runtime.h>

typedef float v2f __attribute__((ext_vector_type(2)));
typedef float v4f __attribute__((ext_vector_type(4)));
typedef float v8f __attribute__((ext_vector_type(8)));

#define B_   2
#define N_   9
#define CB_  80
#define H_   120
#define W_   160
#define HW_  (H_*W_)
#define D_   64
#define P_   80000
#define NEGV (-10000.0f)

// -----------------------------------------------------------------------------
// Kernel A: key_maps[bn, pix, d] = sum_c IF[bn, c, pix] * Wk[d, c] + bk[d]
// WMMA f32 16x16x4: M = d (4 tiles of 16 -> D=64), N = 16 pixels, K = c (step 4)
// A-frag (16x4): lane l (m = l&15, koff = 2*(l>>4)) holds Wk[dbase+m][k0+koff..+1]
// B-frag (4x16): lane l (n = l&15) holds IF[c=k0+koff..+1][pixbase+n]
// D (16x16):     lane l, vgpr v -> d = dbase + v + 8*(l>>4), pix = pixbase + (l&15)
// -----------------------------------------------------------------------------
__global__ __launch_bounds__(256) void keymap_gemm(
    const float* __restrict__ IF, const float* __restrict__ Wk,
    const float* __restrict__ bk, float* __restrict__ KM)
{
    __shared__ float sW[D_ * CB_];
    __shared__ float sb[D_];
    for (int i = threadIdx.x; i < D_ * CB_; i += 256) sW[i] = Wk[i];
    if (threadIdx.x < D_) sb[threadIdx.x] = bk[threadIdx.x];
    __syncthreads();

    const int lane = threadIdx.x & 31;
    const int wid  = threadIdx.x >> 5;
    const int tile = blockIdx.x * 8 + wid;       // global 16-pixel tile
    const int tilesPerImg = HW_ / 16;            // 1200
    const int bn = tile / tilesPerImg;
    const int pixbase = (tile - bn * tilesPerImg) * 16;
    const int l15  = lane & 15;
    const int hf   = lane >> 4;
    const int koff = hf * 2;

    const float* ifb = IF + (size_t)bn * CB_ * HW_ + pixbase + l15;

    v8f acc[4] = {};
    for (int k0 = 0; k0 < CB_; k0 += 4) {
        v2f bf;
        bf.x = ifb[(size_t)(k0 + koff) * HW_];
        bf.y = ifb[(size_t)(k0 + koff + 1) * HW_];
#pragma unroll
        for (int t = 0; t < 4; ++t) {
            v2f af = *(const v2f*)&sW[(t * 16 + l15) * CB_ + k0 + koff];
            acc[t] = __builtin_amdgcn_wmma_f32_16x16x4_f32(
                false, af, false, bf, (short)0, acc[t], false, false);
        }
    }

    float* ob = KM + ((size_t)bn * HW_ + pixbase + l15) * D_;
#pragma unroll
    for (int t = 0; t < 4; ++t) {
        const int d0 = t * 16 + hf * 8;
        v4f lo, hi;
        lo.x = acc[t][0] + sb[d0 + 0]; lo.y = acc[t][1] + sb[d0 + 1];
        lo.z = acc[t][2] + sb[d0 + 2]; lo.w = acc[t][3] + sb[d0 + 3];
        hi.x = acc[t][4] + sb[d0 + 4]; hi.y = acc[t][5] + sb[d0 + 5];
        hi.z = acc[t][6] + sb[d0 + 6]; hi.w = acc[t][7] + sb[d0 + 7];
        *(v4f*)(ob + d0)     = lo;
        *(v4f*)(ob + d0 + 4) = hi;
    }
}

// -----------------------------------------------------------------------------
// Kernel C: q[p, d] = sum_c F[p, c] * Wq[d, c] + bq[d]   (K = 64)
// Same tiling as keymap_gemm; B-fragment is a contiguous b64 from F's row.
// -----------------------------------------------------------------------------
__global__ __launch_bounds__(256) void q_gemm(
    const float* __restrict__ F, const float* __restrict__ Wq,
    const float* __restrict__ bq, float* __restrict__ Qo)
{
    __shared__ float sW[D_ * D_];
    __shared__ float sb[D_];
    for (int i = threadIdx.x; i < D_ * D_; i += 256) sW[i] = Wq[i];
    if (threadIdx.x < D_) sb[threadIdx.x] = bq[threadIdx.x];
    __syncthreads();

    const int lane = threadIdx.x & 31;
    const int wid  = threadIdx.x >> 5;
    const int tile = blockIdx.x * 8 + wid;
    const int pbase = tile * 16;
    const int l15  = lane & 15;
    const int hf   = lane >> 4;
    const int koff = hf * 2;

    const float* fb = F + (size_t)(pbase + l15) * D_;

    v8f acc[4] = {};
    for (int k0 = 0; k0 < D_; k0 += 4) {
        v2f bf = *(const v2f*)(fb + k0 + koff);
#pragma unroll
        for (int t = 0; t < 4; ++t) {
            v2f af = *(const v2f*)&sW[(t * 16 + l15) * D_ + k0 + koff];
            acc[t] = __builtin_amdgcn_wmma_f32_16x16x4_f32(
                false, af, false, bf, (short)0, acc[t], false, false);
        }
    }

    float* ob = Qo + (size_t)(pbase + l15) * D_;
#pragma unroll
    for (int t = 0; t < 4; ++t) {
        const int d0 = t * 16 + hf * 8;
        v4f lo, hi;
        lo.x = acc[t][0] + sb[d0 + 0]; lo.y = acc[t][1] + sb[d0 + 1];
        lo.z = acc[t][2] + sb[d0 + 2]; lo.w = acc[t][3] + sb[d0 + 3];
        hi.x = acc[t][4] + sb[d0 + 4]; hi.y = acc[t][5] + sb[d0 + 5];
        hi.z = acc[t][6] + sb[d0 + 6]; hi.w = acc[t][7] + sb[d0 + 7];
        *(v4f*)(ob + d0)     = lo;
        *(v4f*)(ob + d0 + 4) = hi;
    }
}

// -----------------------------------------------------------------------------
// Kernel B: per-voxel projection + bilinear gather + 9-way softmax attention.
// One voxel per wave; lane owns feature components d = lane and d = lane + 32.
// -----------------------------------------------------------------------------
__global__ __launch_bounds__(256) void mv_attn(
    const float* __restrict__ KM, const float* __restrict__ Q,
    const float* __restrict__ F,  const int* __restrict__ C4,
    const int* __restrict__ CMask, const float* __restrict__ PR,
    const float* __restrict__ ORG, float* __restrict__ Out)
{
    const int lane = threadIdx.x & 31;
    const int wid  = threadIdx.x >> 5;
    const int p    = blockIdx.x * 8 + wid;

    const int cx = C4[p * 4 + 0], cy = C4[p * 4 + 1];
    const int cz = C4[p * 4 + 2], bi = C4[p * 4 + 3];
    const bool sel = CMask[p] > 1;
    const float wx = (float)cx * 0.16f + ORG[bi * 3 + 0];
    const float wy = (float)cy * 0.16f + ORG[bi * 3 + 1];
    const float wz = (float)cz * 0.16f + ORG[bi * 3 + 2];

    const float qlo = Q[(size_t)p * D_ + lane];
    const float qhi = Q[(size_t)p * D_ + 32 + lane];

    float kflo[N_], kfhi[N_], lg[N_];

#pragma unroll
    for (int n = 0; n < N_; ++n) {
        const float* pr = PR + (size_t)(bi * N_ + n) * 12;
        const float c0 = pr[0] * wx + pr[1] * wy + pr[2]  * wz + pr[3];
        const float c1 = pr[4] * wx + pr[5] * wy + pr[6]  * wz + pr[7];
        const float z  = pr[8] * wx + pr[9] * wy + pr[10] * wz + pr[11];
        const float gx = 2.0f * (c0 / z) * (1.0f / (W_ - 1)) - 1.0f;
        const float gy = 2.0f * (c1 / z) * (1.0f / (H_ - 1)) - 1.0f;
        const bool  m  = (fabsf(gx) <= 1.0f) && (fabsf(gy) <= 1.0f) && (z > 0.0f);

        float slo = 0.0f, shi = 0.0f;
        if (m) {                      // wave-uniform branch
            const float px = (gx + 1.0f) * 0.5f * (W_ - 1);
            const float py = (gy + 1.0f) * 0.5f * (H_ - 1);
            const float x0f = floorf(px), y0f = floorf(py);
            const float fx = px - x0f, fy = py - y0f;
            const int x0 = (int)x0f, y0 = (int)y0f;
            const float w00 = (1.0f - fx) * (1.0f - fy), w01 = fx * (1.0f - fy);
            const float w10 = (1.0f - fx) * fy,          w11 = fx * fy;
            const float wgts[4] = {w00, w01, w10, w11};
            const float* kb = KM + (size_t)(bi * N_ + n) * HW_ * D_;
#pragma unroll
            for (int tap = 0; tap < 4; ++tap) {
                const int dy = tap >> 1, dx = tap & 1;
                const int xi = x0 + dx, yi = y0 + dy;
                const bool valid = (xi >= 0) && (xi < W_) && (yi >= 0) && (yi < H_);
                const int xc = xi < 0 ? 0 : (xi > W_ - 1 ? W_ - 1 : xi);
                const int yc = yi < 0 ? 0 : (yi > H_ - 1 ? H_ - 1 : yi);
                const float wgt = valid ? wgts[tap] : 0.0f;
                const float* v = kb + (size_t)(yc * W_ + xc) * D_;
                slo += wgt * v[lane];
                shi += wgt * v[32 + lane];
            }
        }
        kflo[n] = slo;
        kfhi[n] = shi;

        float part = qlo * slo + qhi * shi;
#pragma unroll
        for (int off = 16; off >= 1; off >>= 1)
            part += __shfl_xor(part, off, 32);
        lg[n] = m ? part * 0.125f : NEGV;     // 1/sqrt(64)
    }

    float mx = lg[0];
#pragma unroll
    for (int n = 1; n < N_; ++n) mx = fmaxf(mx, lg[n]);
    float e[N_], s = 0.0f;
#pragma unroll
    for (int n = 0; n < N_; ++n) { e[n] = __expf(lg[n] - mx); s += e[n]; }
    const float inv = 1.0f / s;
    float ylo = 0.0f, yhi = 0.0f;
#pragma unroll
    for (int n = 0; n < N_; ++n) { ylo += e[n] * kflo[n]; yhi += e[n] * kfhi[n]; }
    ylo *= inv; yhi *= inv;

    const float flo = F[(size_t)p * D_ + lane];
    const float fhi = F[(size_t)p * D_ + 32 + lane];
    Out[(size_t)p * D_ + lane]      = flo + (sel ? ylo : 0.0f);
    Out[(size_t)p * D_ + 32 + lane] = fhi + (sel ? yhi : 0.0f);
}

// -----------------------------------------------------------------------------
extern "C" void kernel_launch(void* const* d_in, const int* in_sizes, int n_in,
                              void* d_out, int out_size, void* d_ws, size_t ws_size,
                              hipStream_t stream)
{
    (void)in_sizes; (void)n_in; (void)out_size; (void)ws_size;
    const float* IF  = (const float*)d_in[0];
    const int*   C4  = (const int*)  d_in[1];
    const float* VF  = (const float*)d_in[2];
    const float* PR  = (const float*)d_in[3];
    const float* ORG = (const float*)d_in[4];
    const int*   CM  = (const int*)  d_in[5];
    const float* Wq1 = (const float*)d_in[6];
    const float* bq1 = (const float*)d_in[7];
    const float* Wk1 = (const float*)d_in[8];
    const float* bk1 = (const float*)d_in[9];
    const float* Wq2 = (const float*)d_in[10];
    const float* bq2 = (const float*)d_in[11];
    const float* Wk2 = (const float*)d_in[12];
    const float* bk2 = (const float*)d_in[13];

    float* out = (float*)d_out;
    float* km  = (float*)d_ws;                        // 88.5 MB key maps
    float* qb  = km + (size_t)B_ * N_ * HW_ * D_;     // 20.5 MB queries

    const dim3 blk(256);
    const int kmBlocks = (B_ * N_ * HW_ / 16) / 8;    // 2700 (exact)
    const int qBlocks  = (P_ / 16) / 8;               // 625  (exact)
    const int aBlocks  = P_ / 8;                      // 10000 (exact)

    // ---- block 1 ----
    keymap_gemm<<<kmBlocks, blk, 0, stream>>>(IF, Wk1, bk1, km);
    q_gemm<<<qBlocks, blk, 0, stream>>>(VF, Wq1, bq1, qb);
    mv_attn<<<aBlocks, blk, 0, stream>>>(km, qb, VF, C4, CM, PR, ORG, out);
    // ---- block 2 (in-place on d_out: each thread reads its row then writes it) ----
    keymap_gemm<<<kmBlocks, blk, 0, stream>>>(IF, Wk2, bk2, km);
    q_gemm<<<qBlocks, blk, 0, stream>>>(out, Wq2, bq2, qb);
    mv_attn<<<aBlocks, blk, 0, stream>>>(km, qb, out, C4, CM, PR, ORG, out);
}